// LossWassersteinFull_7627861918340
// MI455X (gfx1250) — compile-verified
//
#include <hip/hip_runtime.h>
#include <math.h>

// ---------------------------------------------------------------------------
// Sinkhorn divergence (geomloss-style, debiased) for N=M=4096, D=64, fp32.
// Cost matrices built with V_WMMA_F32_16X16X4_F32; annealing loop is a
// bandwidth-bound streaming logsumexp over L2-resident 64MB matrices.
// ---------------------------------------------------------------------------

typedef __attribute__((ext_vector_type(2))) float v2f;
typedef __attribute__((ext_vector_type(8))) float v8f;

static constexpr int    NPTS  = 4096;
static constexpr int    DIM   = 64;
static constexpr int    MAX_ITERS = 56;            // worst-case annealing steps
static constexpr size_t MATF  = (size_t)NPTS * NPTS;     // floats per matrix
// ws layout (float offsets)
static constexpr size_t OFF_CXY  = 0;
static constexpr size_t OFF_CYX  = 1 * MATF;
static constexpr size_t OFF_CXX  = 2 * MATF;
static constexpr size_t OFF_CYY  = 3 * MATF;
static constexpr size_t OFF_POT  = 4 * MATF;             // [2][4][NPTS] ping-pong potentials
static constexpr size_t OFF_FIN  = OFF_POT + 2 * 4 * (size_t)NPTS;  // [4][NPTS]
static constexpr size_t OFF_X2   = OFF_FIN + 4 * (size_t)NPTS;
static constexpr size_t OFF_Y2   = OFF_X2  + (size_t)NPTS;
static constexpr size_t OFF_EPS  = OFF_Y2  + (size_t)NPTS;          // up to 64 eps values
static constexpr size_t OFF_META = OFF_EPS + 64;                    // int n_eps

#define LOG_NPTS 8.317766166719343f   /* ln(4096) */

// ---------------------------------------------------------------------------
// 1) diameter + epsilon schedule (single block)
// ---------------------------------------------------------------------------
__global__ void sk_setup_schedule(const float* __restrict__ x,
                                  const float* __restrict__ y,
                                  float* __restrict__ ws) {
  __shared__ float smin[1024];
  __shared__ float smax[1024];
  const int t   = threadIdx.x;        // 0..1023
  const int dim = t & 63;
  const int sub = t >> 6;             // 0..15 threads per dim
  float mn = INFINITY, mx = -INFINITY;
  for (int r = sub; r < NPTS; r += 16) {
    float a = x[r * DIM + dim];
    float b = y[r * DIM + dim];
    mn = fminf(mn, fminf(a, b));
    mx = fmaxf(mx, fmaxf(a, b));
  }
  smin[t] = mn; smax[t] = mx;
  __syncthreads();
  for (int off = 512; off >= 64; off >>= 1) {
    if (t < off) {
      smin[t] = fminf(smin[t], smin[t + off]);
      smax[t] = fmaxf(smax[t], smax[t + off]);
    }
    __syncthreads();
  }
  if (t < 64) {
    float d = smax[t] - smin[t];
    smin[t] = d * d;                  // reuse as squared extent per dim
  }
  __syncthreads();
  if (t == 0) {
    float ss = 0.f;
    for (int k = 0; k < 64; ++k) ss += smin[k];
    float diamf = sqrtf(ss);          // float32 norm like np.linalg.norm(f32)
    double d     = (double)diamf;
    double start = 2.0 * log(d);      // P * log(diameter)
    double stop  = 2.0 * log(0.05);   // P * log(BLUR)
    double step  = 2.0 * log(0.8);    // P * log(SCALING)  (negative)
    int len = (int)ceil((stop - start) / step);
    if (len < 0) len = 0;
    int n = len + 2;
    if (n > MAX_ITERS) n = MAX_ITERS;
    float* eps = ws + OFF_EPS;
    eps[0] = (float)(d * d);          // diameter ** P
    int idx = 1;
    for (int k = 0; k < len && idx < n - 1; ++k)
      eps[idx++] = (float)exp(start + (double)k * step);
    eps[n - 1] = 0.0025f;             // BLUR ** P
    ((int*)(ws + OFF_META))[0] = n;
  }
}

// ---------------------------------------------------------------------------
// 2) squared norms of x and y rows
// ---------------------------------------------------------------------------
__global__ void sk_norms(const float* __restrict__ x,
                         const float* __restrict__ y,
                         float* __restrict__ ws) {
  const int i = blockIdx.x * blockDim.x + threadIdx.x;  // 0..8191
  const float* p = (i < NPTS) ? (x + (size_t)i * DIM)
                              : (y + (size_t)(i - NPTS) * DIM);
  float s = 0.f;
#pragma unroll
  for (int k = 0; k < DIM / 4; ++k) {
    float4 q = *(const float4*)(p + 4 * k);
    s += q.x * q.x + q.y * q.y + q.z * q.z + q.w * q.w;
  }
  if (i < NPTS) ws[OFF_X2 + i] = s;
  else          ws[OFF_Y2 + (i - NPTS)] = s;
}

// ---------------------------------------------------------------------------
// 3) cost matrices via fp32 WMMA: C = 0.5*(|a|^2 + |b|^2) - a.b
//    z=0: C_xy (+ explicit transpose into C_yx), z=1: C_xx, z=2: C_yy
//    One wave computes one 16x16 tile; K=64 chained as 16x V_WMMA_F32_16X16X4_F32
// ---------------------------------------------------------------------------
__global__ __launch_bounds__(128) void sk_cost_gemm(const float* __restrict__ x,
                                                    const float* __restrict__ y,
                                                    float* __restrict__ ws) {
  const int z = blockIdx.z;
  const float *A, *B, *nA, *nB;
  float *C, *CT = nullptr;
  if (z == 0)      { A = x; B = y; nA = ws + OFF_X2; nB = ws + OFF_Y2;
                     C = ws + OFF_CXY; CT = ws + OFF_CYX; }
  else if (z == 1) { A = x; B = x; nA = ws + OFF_X2; nB = ws + OFF_X2;
                     C = ws + OFF_CXX; }
  else             { A = y; B = y; nA = ws + OFF_Y2; nB = ws + OFF_Y2;
                     C = ws + OFF_CYY; }

  const int lane = threadIdx.x;                 // 0..31 (wave32)
  const int wv   = threadIdx.y;                 // 0..3 waves per block
  const int j0   = (blockIdx.x * 4 + wv) * 16;  // N tile
  const int i0   = blockIdx.y * 16;             // M tile
  const int lm   = lane & 15;
  const int lh   = lane >> 4;

  // fp32 WMMA A (16x4): lane = M, VGPR0/1 hold K = 2*lh, 2*lh+1
  // fp32 WMMA B (4x16): lane = N, VGPR0/1 hold K = 2*lh, 2*lh+1  -> same float2 load shape
  const float* ap = A + (size_t)(i0 + lm) * DIM + 2 * lh;
  const float* bp = B + (size_t)(j0 + lm) * DIM + 2 * lh;

  v8f acc = {};
#pragma unroll
  for (int k0 = 0; k0 < DIM; k0 += 4) {
    v2f a = *(const v2f*)(ap + k0);
    v2f b = *(const v2f*)(bp + k0);
    acc = __builtin_amdgcn_wmma_f32_16x16x4_f32(
        /*neg_a=*/false, a, /*neg_b=*/false, b,
        /*c_mod=*/(short)0, acc, /*reuse_a=*/false, /*reuse_b=*/false);
  }

  const float bn = nB[j0 + lm];   // column norm (fixed per lane)
#pragma unroll
  for (int r = 0; r < 8; ++r) {
    const int row = i0 + r + 8 * lh;   // C/D layout: VGPR r, lanes 16-31 -> M+8
    const int col = j0 + lm;
    float v = 0.5f * (nA[row] + bn) - acc[r];
    C[(size_t)row * NPTS + col] = v;
    if (CT) CT[(size_t)col * NPTS + row] = v;
  }
}

// ---------------------------------------------------------------------------
// 4) fused softmin update.  grid = (4096 rows, 4 updates)
//    w=0: f_ba over C_xy with pot g_ab   w=1: g_ab over C_yx with pot f_ba
//    w=2: f_aa over C_xx with pot f_aa   w=3: g_bb over C_yy with pot g_bb
//    mode 0: init (pot=0, write parity 0)
//    mode 1: iterate (ping-pong, out = 0.5*(old+ft); skip-copy if iter>=n_eps)
//    mode 2: final extrapolation (out -> fin[])
// ---------------------------------------------------------------------------
__global__ __launch_bounds__(256) void sk_softmin(float* __restrict__ ws,
                                                  int mode, int iter) {
  const int row = blockIdx.x;
  const int w   = blockIdx.y;
  const int t   = threadIdx.x;

  const int neps = ((const int*)(ws + OFF_META))[0];
  float* pot = ws + OFF_POT;                        // [2][4][NPTS]
  const int potw = (w == 0) ? 1 : (w == 1) ? 0 : w; // which potential feeds update w
  const float* C = ws + (size_t)w * MATF;           // CXY,CYX,CXX,CYY in order

  float eps;
  const float* potp;     // nullptr => zero potential
  float* outp;
  int rp = 0;
  if (mode == 0) {
    eps  = ws[OFF_EPS];
    potp = nullptr;
    outp = pot + (size_t)(0 * 4 + w) * NPTS;
  } else if (mode == 1) {
    rp = iter & 1;
    const int wp = (iter + 1) & 1;
    if (iter >= neps) {                              // schedule exhausted: copy state
      if (t == 0)
        pot[(size_t)(wp * 4 + w) * NPTS + row] = pot[(size_t)(rp * 4 + w) * NPTS + row];
      return;
    }
    eps  = ws[OFF_EPS + iter];
    potp = pot + (size_t)(rp * 4 + potw) * NPTS;
    outp = pot + (size_t)(wp * 4 + w) * NPTS;
  } else {                                           // final extrapolation
    rp   = MAX_ITERS & 1;                            // parity after fixed host loop
    eps  = ws[OFF_EPS + neps - 1];                   // == BLUR**P
    potp = pot + (size_t)(rp * 4 + potw) * NPTS;
    outp = ws + OFF_FIN + (size_t)w * NPTS;
  }

  const float inv = 1.0f / eps;
  const float* Crow = C + (size_t)row * NPTS;

  // 16 elements per thread: v_j = (pot_j - C_ij) / eps ; streaming max then sum
  float v[16];
  float m = -INFINITY;
#pragma unroll
  for (int k = 0; k < 4; ++k) {
    const int j = 4 * t + 1024 * k;
    float4 c4 = *(const float4*)(Crow + j);
    float4 p4 = potp ? *(const float4*)(potp + j) : make_float4(0.f, 0.f, 0.f, 0.f);
    v[4 * k + 0] = (p4.x - c4.x) * inv;
    v[4 * k + 1] = (p4.y - c4.y) * inv;
    v[4 * k + 2] = (p4.z - c4.z) * inv;
    v[4 * k + 3] = (p4.w - c4.w) * inv;
    m = fmaxf(m, fmaxf(fmaxf(v[4 * k + 0], v[4 * k + 1]),
                       fmaxf(v[4 * k + 2], v[4 * k + 3])));
  }

  __shared__ float red[256];
  red[t] = m;
  __syncthreads();
  for (int off = 128; off >= 1; off >>= 1) {
    if (t < off) red[t] = fmaxf(red[t], red[t + off]);
    __syncthreads();
  }
  const float M = red[0];
  __syncthreads();

  float s = 0.f;
#pragma unroll
  for (int k = 0; k < 16; ++k) s += __expf(v[k] - M);
  red[t] = s;
  __syncthreads();
  for (int off = 128; off >= 1; off >>= 1) {
    if (t < off) red[t] += red[t + off];
    __syncthreads();
  }

  if (t == 0) {
    // softmin = -eps * (-lnN + M + ln S) = eps * (lnN - M - ln S)
    float ft = eps * (LOG_NPTS - M - __logf(red[0]));
    if (mode == 1)
      ft = 0.5f * (pot[(size_t)(rp * 4 + w) * NPTS + row] + ft);
    outp[row] = ft;
  }
}

// ---------------------------------------------------------------------------
// 5) result = mean(f_ba - f_aa) + mean(g_ab - g_bb)
// ---------------------------------------------------------------------------
__global__ void sk_reduce(const float* __restrict__ ws, float* __restrict__ out) {
  const float* fin = ws + OFF_FIN;   // [4][NPTS]: f_ba, g_ab, f_aa, g_bb
  const int t = threadIdx.x;         // 256 threads
  float s = 0.f;
  for (int k = 0; k < 64; ++k) {
    const int idx = t + 256 * k;     // 0..16383
    const int w = idx >> 12;
    s += (w < 2) ? fin[idx] : -fin[idx];
  }
  __shared__ float red[256];
  red[t] = s;
  __syncthreads();
  for (int off = 128; off >= 1; off >>= 1) {
    if (t < off) red[t] += red[t + off];
    __syncthreads();
  }
  if (t == 0) out[0] = red[0] * (1.0f / (float)NPTS);
}

// ---------------------------------------------------------------------------
extern "C" void kernel_launch(void* const* d_in, const int* in_sizes, int n_in,
                              void* d_out, int out_size, void* d_ws, size_t ws_size,
                              hipStream_t stream) {
  (void)in_sizes; (void)n_in; (void)out_size; (void)ws_size;
  const float* x = (const float*)d_in[0];
  const float* y = (const float*)d_in[1];
  float* ws  = (float*)d_ws;
  float* out = (float*)d_out;

  sk_setup_schedule<<<1, 1024, 0, stream>>>(x, y, ws);
  sk_norms<<<(2 * NPTS) / 256, 256, 0, stream>>>(x, y, ws);
  sk_cost_gemm<<<dim3(NPTS / 64, NPTS / 16, 3), dim3(32, 4), 0, stream>>>(x, y, ws);

  sk_softmin<<<dim3(NPTS, 4), 256, 0, stream>>>(ws, 0, 0);          // init at eps0
  for (int it = 0; it < MAX_ITERS; ++it)                            // fixed worst-case loop
    sk_softmin<<<dim3(NPTS, 4), 256, 0, stream>>>(ws, 1, it);
  sk_softmin<<<dim3(NPTS, 4), 256, 0, stream>>>(ws, 2, 0);          // final extrapolation

  sk_reduce<<<1, 256, 0, stream>>>(ws, out);
}